// Normalized_Correlation_Layer_88252987998983
// MI455X (gfx1250) — compile-verified
//
#include <hip/hip_runtime.h>

// ---- problem constants (from reference) ----
#define K0_   12
#define K1_   5
#define PATCH 60            // K0*K1
#define KPAD  64            // padded K (zeros in 60..63)
#define B_    8
#define H_    48
#define W_    16
#define C_    5
#define R_    37            // H-K0+1
#define CO_   12            // W-K1+1
#define NPATCH 2220         // C*R*CO
#define NPAD   2240         // padded N, multiple of 64
#define LDS_S  68           // LDS row stride in floats (bank-conflict-free)

typedef __attribute__((ext_vector_type(2))) float v2f;
typedef __attribute__((ext_vector_type(8))) float v8f;

// -------------------------------------------------------------------
// Kernel 1: extract + normalize patches into padded [b][n][KPAD] layout
// -------------------------------------------------------------------
__global__ __launch_bounds__(256) void nc_prep(const float* __restrict__ x,
                                               float* __restrict__ ws) {
    int gid = blockIdx.x * 256 + threadIdx.x;
    if (gid >= B_ * NPAD) return;
    int b = gid / NPAD;
    int n = gid % NPAD;
    float* dst = ws + (size_t)gid * KPAD;
    if (n >= NPATCH) {                // zero-pad rows beyond 2220
        #pragma unroll
        for (int k = 0; k < KPAD; k += 4)
            *(float4*)(dst + k) = float4{0.f, 0.f, 0.f, 0.f};
        return;
    }
    int c  = n / (R_ * CO_);
    int rc = n % (R_ * CO_);
    int r  = rc / CO_;
    int co = rc % CO_;
    const float* base = x + ((size_t)(b * H_ + r) * W_ + co) * C_ + c;

    float v[PATCH];
    float s = 0.f, ss = 0.f;
    #pragma unroll
    for (int i = 0; i < K0_; ++i) {
        #pragma unroll
        for (int j = 0; j < K1_; ++j) {
            float t = base[(i * W_ + j) * C_];
            v[i * K1_ + j] = t;
            s  += t;
            ss += t * t;
        }
    }
    float mean = s * (1.0f / PATCH);
    float var  = ss * (1.0f / PATCH) - mean * mean;
    float inv  = 1.0f / sqrtf(var);
    #pragma unroll
    for (int k = 0; k < PATCH; ++k)
        dst[k] = (v[k] - mean) * inv;
    dst[60] = 0.f; dst[61] = 0.f; dst[62] = 0.f; dst[63] = 0.f;
}

// -------------------------------------------------------------------
// Kernel 2: out[b] = A @ B^T via V_WMMA_F32_16X16X4_F32
//   A = normalized patches of input_2  (rows of output, einsum 'n')
//   B = normalized patches of input_1  (cols of output, einsum 'm')
// block = 256 threads (8 waves) -> 64(M) x 64(N) output block.
// Each wave computes 16x32 (two 16x16 WMMA tiles sharing the A fragment),
// K = 60 in 15 steps of 4.
// -------------------------------------------------------------------
__global__ __launch_bounds__(256) void nc_gemm(const float* __restrict__ Aws,
                                               const float* __restrict__ Bws,
                                               float* __restrict__ out) {
    __shared__ float lA[64 * LDS_S];
    __shared__ float lB[64 * LDS_S];

    const int b   = blockIdx.z;
    const int m0  = blockIdx.x * 64;
    const int n0  = blockIdx.y * 64;
    const int tid = threadIdx.x;

    // cooperative LDS fill: 128 rows x 64 floats, float4-coalesced
    const float* gA = Aws + ((size_t)b * NPAD + m0) * KPAD;
    const float* gB = Bws + ((size_t)b * NPAD + n0) * KPAD;
    #pragma unroll
    for (int i = 0; i < 8; ++i) {
        int id  = tid + i * 256;
        int row = id >> 4;          // 0..127
        int j   = (id & 15) * 4;    // 0..60
        if (row < 64) {
            float4 val = *(const float4*)(gA + (size_t)row * KPAD + j);
            *(float4*)(lA + row * LDS_S + j) = val;
        } else {
            float4 val = *(const float4*)(gB + (size_t)(row - 64) * KPAD + j);
            *(float4*)(lB + (row - 64) * LDS_S + j) = val;
        }
    }
    __syncthreads();

    const int wave = tid >> 5;      // 0..7
    const int lane = tid & 31;
    const int tm   = wave & 3;      // M tile within block: 0..3  (16 rows each)
    const int tn   = wave >> 2;     // N half within block: 0..1  (32 cols each)
    const int rloc = lane & 15;     // M (for A) / N (for B) within tile
    const int hi   = lane >> 4;     // K half-select per ISA f32 A/B layout

    const float* aRow  = lA + (tm * 16 + rloc) * LDS_S;
    const float* bRow0 = lB + (tn * 32 + rloc) * LDS_S;
    const float* bRow1 = bRow0 + 16 * LDS_S;

    v8f acc0 = {}, acc1 = {};
    #pragma unroll
    for (int k = 0; k < PATCH; k += 4) {
        int kk  = k + 2 * hi;
        v2f a   = *(const v2f*)(aRow  + kk);  // lanes0-15 K={k,k+1}, lanes16-31 K={k+2,k+3}
        v2f bv0 = *(const v2f*)(bRow0 + kk);
        v2f bv1 = *(const v2f*)(bRow1 + kk);
        acc0 = __builtin_amdgcn_wmma_f32_16x16x4_f32(false, a, false, bv0,
                                                     (short)0, acc0, false, false);
        acc1 = __builtin_amdgcn_wmma_f32_16x16x4_f32(false, a, false, bv1,
                                                     (short)0, acc1, false, false);
    }

    // D layout: VGPR g, lanes0-15 -> M=g, lanes16-31 -> M=g+8; N=lane%16
    const int rowBase = m0 + tm * 16 + 8 * hi;
    const int col0    = n0 + tn * 32 + rloc;
    const int col1    = col0 + 16;
    float* outB = out + (size_t)b * NPATCH * NPATCH;

    if (m0 + 64 <= NPATCH && n0 + 64 <= NPATCH) {
        // interior fast path: uniform branch, straight-line coalesced stores
        #pragma unroll
        for (int g = 0; g < 8; ++g) {
            float* rp = outB + (size_t)(rowBase + g) * NPATCH;
            rp[col0] = acc0[g];
            rp[col1] = acc1[g];
        }
    } else {
        #pragma unroll
        for (int g = 0; g < 8; ++g) {
            int row = rowBase + g;
            if (row < NPATCH) {
                float* rp = outB + (size_t)row * NPATCH;
                if (col0 < NPATCH) rp[col0] = acc0[g];
                if (col1 < NPATCH) rp[col1] = acc1[g];
            }
        }
    }
}

// -------------------------------------------------------------------
extern "C" void kernel_launch(void* const* d_in, const int* in_sizes, int n_in,
                              void* d_out, int out_size, void* d_ws, size_t ws_size,
                              hipStream_t stream) {
    const float* in1 = (const float*)d_in[0];   // input_1 -> B side (cols, 'm')
    const float* in2 = (const float*)d_in[1];   // input_2 -> A side (rows, 'n')
    float* Aws = (float*)d_ws;                              // 8*2240*64 floats
    float* Bws = Aws + (size_t)B_ * NPAD * KPAD;            // 8*2240*64 floats
    // total ws use: ~9.2 MB (L2-resident)

    int nthreads = B_ * NPAD;
    int nblocks  = (nthreads + 255) / 256;
    nc_prep<<<nblocks, 256, 0, stream>>>(in2, Aws);
    nc_prep<<<nblocks, 256, 0, stream>>>(in1, Bws);

    dim3 grid(NPAD / 64, NPAD / 64, B_);
    nc_gemm<<<grid, 256, 0, stream>>>(Aws, Bws, (float*)d_out);
}